// BlobSplatter_46892452938081
// MI455X (gfx1250) — compile-verified
//
#include <hip/hip_runtime.h>
#include <hip/hip_bf16.h>
#include <math.h>

typedef __attribute__((ext_vector_type(2))) float v2f;
typedef __attribute__((ext_vector_type(8))) float v8f;

#define NB 8
#define BATCH 256
#define H 64
#define TS 256
#define LSTRIDE 65   // pad 64->65 floats to avoid LDS bank conflicts

__device__ __constant__ int   c_side[NB] = {1,0,1,0,1,0,1,0};
__device__ __constant__ float c_sy[NB]   = {0.1f,0.2f,0.3f,0.4f,0.5f,0.6f,0.7f,0.8f};
__device__ __constant__ float c_sx[NB]   = {0.8f,0.7f,0.6f,0.5f,0.4f,0.3f,0.2f,0.1f};

__device__ __forceinline__ float fast_sigmoid(float x) {
    // 1 / (1 + e^-x) = rcp(1 + exp2(-log2e * x))
    float e = __builtin_amdgcn_exp2f(-1.44269504088896f * x);
    return __builtin_amdgcn_rcpf(1.0f + e);
}

// ---------------------------------------------------------------------------
// Phase 1: per-blob MLP encode via V_WMMA_F32_16X16X4_F32.
// One block per blob. 256 threads = 8 wave32. Each wave owns 8 of the 64
// 16x16 tiles of the [256 x 64] activation matrix (2 M-tiles x 4 N-tiles).
// ---------------------------------------------------------------------------
__global__ __launch_bounds__(256) void blob_encode_kernel(
    const float* __restrict__ pos,   // [256,6]
    const float* __restrict__ W1,    // [8,3,64]
    const float* __restrict__ b1,    // [8,64]
    const float* __restrict__ W2,    // [8,64,64]
    const float* __restrict__ b2,    // [8,64]
    const float* __restrict__ W3,    // [8,64,5]
    const float* __restrict__ b3,    // [8,5]
    const float* __restrict__ scale_p,
    float* __restrict__ P)           // [8,256,6] -> {y,x,cos,sin,inv_sa,inv_sb}
{
    const int blob = blockIdx.x;
    const int tid  = threadIdx.x;
    const int lane = tid & 31;
    const int wave = tid >> 5;
    const int l15  = lane & 15;
    const int lhi  = lane >> 4;          // 0: K even pair base, 1: K odd half

    __shared__ float h1[BATCH * LSTRIDE];
    __shared__ float h2[BATCH * LSTRIDE];

    // ---------- Layer 1: relu(100*pos_sel @ W1 + b1)  (K=3 padded to 4) ----
    const int poff = c_side[blob] ? 0 : 3;
    #pragma unroll
    for (int i = 0; i < 8; ++i) {
        const int mt = wave * 2 + (i >> 2);
        const int nt = i & 3;
        const int m  = mt * 16 + l15;    // A row (lanes 0-15 and 16-31 share M)
        const int n  = nt * 16 + l15;    // B col
        const int k0 = lhi ? 2 : 0;
        const int k1 = lhi ? 3 : 1;
        v2f A, Bm;
        A.x  = pos[m * 6 + poff + k0] * 100.0f;          // k0 in {0,2}
        A.y  = (k1 < 3) ? pos[m * 6 + poff + k1] * 100.0f : 0.0f;
        Bm.x = W1[blob * 3 * H + k0 * H + n];
        Bm.y = (k1 < 3) ? W1[blob * 3 * H + k1 * H + n] : 0.0f;
        v8f C = {};
        C = __builtin_amdgcn_wmma_f32_16x16x4_f32(false, A, false, Bm,
                                                  (short)0, C, false, false);
        const float bias = b1[blob * H + n];
        #pragma unroll
        for (int r = 0; r < 8; ++r) {
            float v = C[r] + bias;
            v = v > 0.0f ? v : 0.0f;
            const int mrow = mt * 16 + r + (lhi ? 8 : 0);   // C/D row layout
            h1[mrow * LSTRIDE + n] = v;
        }
    }
    __syncthreads();

    // ---------- Layer 2: relu(h1 @ W2 + b2), K=64 -> 16 chained WMMAs ------
    #pragma unroll
    for (int i = 0; i < 8; ++i) {
        const int mt = wave * 2 + (i >> 2);
        const int nt = i & 3;
        const int m  = mt * 16 + l15;
        const int n  = nt * 16 + l15;
        v8f C = {};
        #pragma unroll
        for (int k = 0; k < H; k += 4) {
            const int k0 = k + (lhi ? 2 : 0);
            const int k1 = k + (lhi ? 3 : 1);
            v2f A, Bm;
            A.x  = h1[m * LSTRIDE + k0];
            A.y  = h1[m * LSTRIDE + k1];
            Bm.x = W2[blob * H * H + k0 * H + n];
            Bm.y = W2[blob * H * H + k1 * H + n];
            C = __builtin_amdgcn_wmma_f32_16x16x4_f32(false, A, false, Bm,
                                                      (short)0, C, false, false);
        }
        const float bias = b2[blob * H + n];
        #pragma unroll
        for (int r = 0; r < 8; ++r) {
            float v = C[r] + bias;
            v = v > 0.0f ? v : 0.0f;
            h2[(mt * 16 + r + (lhi ? 8 : 0)) * LSTRIDE + n] = v;
        }
    }
    __syncthreads();

    // ---------- Layer 3 (N=5, scalar) + parameter post-processing ----------
    {
        const int b = tid;   // one batch row per thread
        float a0 = b3[blob * 5 + 0], a1 = b3[blob * 5 + 1], a2 = b3[blob * 5 + 2];
        float a3 = b3[blob * 5 + 3], a4 = b3[blob * 5 + 4];
        const float* w3 = W3 + blob * H * 5;
        #pragma unroll 8
        for (int k = 0; k < H; ++k) {
            const float hv = h2[b * LSTRIDE + k];
            a0 = fmaf(hv, w3[k * 5 + 0], a0);
            a1 = fmaf(hv, w3[k * 5 + 1], a1);
            a2 = fmaf(hv, w3[k * 5 + 2], a2);
            a3 = fmaf(hv, w3[k * 5 + 3], a3);
            a4 = fmaf(hv, w3[k * 5 + 4], a4);
        }
        const float scale = scale_p[0];
        const float y  = fast_sigmoid(a0) + c_sy[blob];
        const float x  = fast_sigmoid(a1) + c_sx[blob];
        const float s  = (a2 + 0.05f) * scale;
        const float av = fmaf(fast_sigmoid(a3), 1.5f, 0.5f);      // [0.5, 2.0]
        const float th = fast_sigmoid(a4) * 3.14159265358979f;
        const float ct = cosf(th);
        const float st = sinf(th);
        const float inv_sa = __builtin_amdgcn_rcpf(fmaf(s, av, 1e-6f));
        const float sb = fmaf(s, __builtin_amdgcn_rcpf(av + 1e-6f), 1e-6f);
        const float inv_sb = __builtin_amdgcn_rcpf(sb);
        float* Pp = P + (blob * BATCH + b) * 6;
        Pp[0] = y;  Pp[1] = x;  Pp[2] = ct;
        Pp[3] = st; Pp[4] = inv_sa; Pp[5] = inv_sb;
    }
}

// ---------------------------------------------------------------------------
// Phase 2: splat. grid = (32 row-chunks, 256 batches), 256 threads = one
// column each, 8 rows per thread. Column-invariant per-blob terms hoisted.
// Inner loop per blob: 3 FMA + 2 MUL + v_exp_f32 + accumulate FMA.
// ---------------------------------------------------------------------------
__global__ __launch_bounds__(256) void blob_splat_kernel(
    const float* __restrict__ P,     // [8,256,6]
    float* __restrict__ out)         // [256,256,256]
{
    __shared__ float p[NB * 6];
    const int b   = blockIdx.y;
    const int tid = threadIdx.x;
    if (tid < NB * 6) {
        const int blob = tid / 6, j = tid - blob * 6;
        p[tid] = P[(blob * BATCH + b) * 6 + j];
    }
    __syncthreads();

    const float cx = (tid + 0.5f) * (1.0f / TS);
    float py[NB], us[NB], vc[NB], ct[NB], st[NB], isa[NB], isb[NB];
    #pragma unroll
    for (int i = 0; i < NB; ++i) {
        const float yy = p[i * 6 + 0];
        const float xx = p[i * 6 + 1];
        const float c  = p[i * 6 + 2];
        const float sn = p[i * 6 + 3];
        isa[i] = p[i * 6 + 4];
        isb[i] = p[i * 6 + 5];
        const float dx = cx - xx;
        py[i] = yy; ct[i] = c; st[i] = sn;
        us[i] = sn * dx;       // u = ct*dy + sn*dx
        vc[i] = c  * dx;       // v = ct*dx - sn*dy
    }

    const int row0 = blockIdx.x * 8;
    float* orow = out + ((size_t)b << 16) + (size_t)row0 * TS + tid;
    #pragma unroll
    for (int r = 0; r < 8; ++r) {
        const float cy = (row0 + r + 0.5f) * (1.0f / TS);
        float acc = 0.0f;
        #pragma unroll
        for (int i = 0; i < NB; ++i) {
            const float dy = cy - py[i];
            const float u  = fmaf(ct[i],  dy, us[i]);
            const float v  = fmaf(-st[i], dy, vc[i]);
            const float ua = u * isa[i];
            const float vb = v * isb[i];
            const float t  = fmaf(ua, ua, vb * vb);
            // exp(-0.5*t) == exp2(-0.5*log2(e)*t)
            const float e  = __builtin_amdgcn_exp2f(-0.72134752044448f * t);
            acc = fmaf(acc, e, e);     // img = img*cur + cur
        }
        orow[(size_t)r * TS] = acc;
    }
}

extern "C" void kernel_launch(void* const* d_in, const int* in_sizes, int n_in,
                              void* d_out, int out_size, void* d_ws, size_t ws_size,
                              hipStream_t stream) {
    const float* pos   = (const float*)d_in[0];
    const float* W1    = (const float*)d_in[1];
    const float* b1    = (const float*)d_in[2];
    const float* W2    = (const float*)d_in[3];
    const float* b2    = (const float*)d_in[4];
    const float* W3    = (const float*)d_in[5];
    const float* b3    = (const float*)d_in[6];
    // d_in[7] = target_size (int, statically 256 per reference)
    const float* scale = (const float*)d_in[8];

    float* params = (float*)d_ws;          // 8*256*6 floats = 48 KB
    float* img    = (float*)d_out;         // 256*256*256 floats

    blob_encode_kernel<<<NB, 256, 0, stream>>>(pos, W1, b1, W2, b2, W3, b3,
                                               scale, params);
    blob_splat_kernel<<<dim3(TS / 8, BATCH), 256, 0, stream>>>(params, img);
}